// ExperimentalLoss_1_23742579212661
// MI455X (gfx1250) — compile-verified
//
#include <hip/hip_runtime.h>

// ---------------------------------------------------------------------------
// Fused masked-MSE + blur/Sobel edge loss, 4096x4096 f32, MI455X (gfx1250).
// Bandwidth-bound: 128 MiB read once -> ~5.8us floor @ 23.3 TB/s.
// One pass: LDS tile of d = p - t (halo 2, reflect-101), LDS tile of
// e = blur3x3(d) (halo 1), Sobel on core, WMMA-based reductions (wave32).
// ---------------------------------------------------------------------------

typedef __attribute__((ext_vector_type(2))) float v2f;
typedef __attribute__((ext_vector_type(8))) float v8f;

#define IMG   4096
#define TW    64
#define THT   64
#define DW    (TW + 4)     // 68: d tile width  (halo 2)
#define DH    (THT + 4)    // 68
#define EW    (TW + 2)     // 66: e tile width  (halo 1)
#define EH    (THT + 2)    // 66
#define NBX   (IMG / TW)   // 64
#define NBY   (IMG / THT)  // 64
#define NBLK  (NBX * NBY)  // 4096

// BORDER_REFLECT_101 index map, valid for g in [-(n-1), 2n-2]
__device__ __forceinline__ int refl(int g, int n) {
    return g < 0 ? -g : (g >= n ? 2 * n - 2 - g : g);
}

// Sum 32 lane-values of a full wave32 via V_WMMA_F32_16X16X4_F32.
// A slots (16x4): lanes carry {v, 0}; B = all-ones => D[i][j] = rowsum_i.
// Per-lane sum of the 8 D regs = half the grand total (rows 0-7 vs 8-15
// split across lane and lane^16); one shfl_xor(16) completes it.
// EXEC must be all ones at the call site (callers guarantee no divergence).
__device__ __forceinline__ float wave_sum_wmma(float v) {
    v2f a; a.x = v;    a.y = 0.0f;
    v2f b; b.x = 1.0f; b.y = 1.0f;
    v8f c = {};
    c = __builtin_amdgcn_wmma_f32_16x16x4_f32(false, a, false, b,
                                              (short)0, c, false, false);
    float s = c[0] + c[1] + c[2] + c[3] + c[4] + c[5] + c[6] + c[7];
    s += __shfl_xor(s, 16, 32);
    return s;
}

__global__ __launch_bounds__(256) void
loss_tile_kernel(const float* __restrict__ pred,
                 const float* __restrict__ targ,
                 float* __restrict__ partials) {
    __shared__ float dext[DH * DW];   // d = p - t, reflect-extended
    __shared__ float eext[EH * EW];   // e = blur3x3(d), reflect-extended
    __shared__ float red[3 * 8];

    const int tid = threadIdx.x;
    const int x0  = blockIdx.x * TW;
    const int y0  = blockIdx.y * THT;

    // ---- Phase 1: load d tile (and accumulate masked-MSE on core pixels) ----
    float mse = 0.0f, cnt = 0.0f;
    for (int idx = tid; idx < DH * DW; idx += 256) {
        const int ly = idx / DW;
        const int lx = idx - ly * DW;
        const int ry = refl(y0 - 2 + ly, IMG);
        const int rx = refl(x0 - 2 + lx, IMG);
        const float p = pred[ry * IMG + rx];
        const float t = targ[ry * IMG + rx];
        dext[idx] = p - t;
        // core pixels are loaded exactly once and are never reflected
        if (ly >= 2 && ly < 2 + THT && lx >= 2 && lx < 2 + TW) {
            if (t > 0.0f) {
                const float df = t - p;
                mse += df * df;
                cnt += 1.0f;
            }
        }
    }
    __syncthreads();

    // ---- Phase 2: e = blur3x3(d) on extended tile, reflect into image ------
    // e at out-of-image coords equals e at the reflected in-image coord
    // (matches the reference's second reflect-pad applied to the blurred map).
    for (int idx = tid; idx < EH * EW; idx += 256) {
        const int ly = idx / EW;
        const int lx = idx - ly * EW;
        const int ry = refl(y0 - 1 + ly, IMG);   // in-image row of this e value
        const int rx = refl(x0 - 1 + lx, IMG);
        const int dy = ry - (y0 - 2);            // local row in dext (>=1)
        const int dxi = rx - (x0 - 2);           // local col in dext (>=1)
        const float* r0 = &dext[(dy - 1) * DW + (dxi - 1)];
        const float* r1 = r0 + DW;
        const float* r2 = r1 + DW;
        // (0.25,0.5,0.25) x (0.25,0.5,0.25); dext already holds reflected
        // neighbors, so taps landing outside the image pick correct values.
        eext[idx] = 0.0625f * (r0[0] + r0[2] + r2[0] + r2[2])
                  + 0.125f  * (r0[1] + r2[1] + r1[0] + r1[2])
                  + 0.25f   *  r1[1];
    }
    __syncthreads();

    // ---- Phase 3: Sobel on 64x64 core, accumulate dx^2 - dy^2 --------------
    float g = 0.0f;
    const int lx  = tid & (TW - 1);   // 0..63 (stride-1 within a wave)
    const int lyb = tid >> 6;         // 0..3
    for (int r = 0; r < THT / 4; ++r) {
        const int cy = lyb + 4 * r;                 // core row 0..63
        const float* e0 = &eext[cy * EW + lx];      // rows cy, cy+1, cy+2
        const float* e1 = e0 + EW;                  // center = e1[1]
        const float* e2 = e1 + EW;
        const float dxv = (e0[2] - e0[0]) + 2.0f * (e1[2] - e1[0]) + (e2[2] - e2[0]);
        const float dyv = (e2[0] - e0[0]) + 2.0f * (e2[1] - e0[1]) + (e2[2] - e0[2]);
        g += dxv * dxv - dyv * dyv;   // reference subtracts the y term
    }

    // ---- Phase 4: reduce (WMMA per wave, then LDS across 8 waves) ----------
    const float wm = wave_sum_wmma(mse);
    const float wc = wave_sum_wmma(cnt);
    const float wg = wave_sum_wmma(g);
    const int wave = tid >> 5;
    if ((tid & 31) == 0) {
        red[wave]      = wm;
        red[8 + wave]  = wc;
        red[16 + wave] = wg;
    }
    __syncthreads();
    if (tid == 0) {
        float sm = 0.0f, sc = 0.0f, sg = 0.0f;
        for (int i = 0; i < 8; ++i) {
            sm += red[i];
            sc += red[8 + i];
            sg += red[16 + i];
        }
        const int bid = blockIdx.y * gridDim.x + blockIdx.x;
        partials[bid]            = sm;
        partials[NBLK + bid]     = sc;
        partials[2 * NBLK + bid] = sg;
    }
}

// One wave reduces 3 x 4096 block partials with WMMA (64 values/instruction)
// and writes the final scalar.
__global__ __launch_bounds__(32) void
loss_finalize_kernel(const float* __restrict__ partials,
                     float* __restrict__ out) {
    const int lane = threadIdx.x;
    v2f b; b.x = 1.0f; b.y = 1.0f;
    float tot[3];
    for (int a = 0; a < 3; ++a) {
        const float* src = partials + a * NBLK;
        v8f c = {};
        for (int t = 0; t < NBLK; t += 64) {
            v2f av;
            av.x = src[t + lane];
            av.y = src[t + 32 + lane];
            c = __builtin_amdgcn_wmma_f32_16x16x4_f32(false, av, false, b,
                                                      (short)0, c, false, false);
        }
        float s = c[0] + c[1] + c[2] + c[3] + c[4] + c[5] + c[6] + c[7];
        s += __shfl_xor(s, 16, 32);
        tot[a] = s;
    }
    if (lane == 0) {
        const float denom = tot[1] > 1.0f ? tot[1] : 1.0f;  // max(n_valid, 1)
        const float mae = tot[0] / denom;
        const float G   = tot[2] / ((float)IMG * (float)IMG);
        out[0] = 0.2f * G + 0.8f * mae;
    }
}

extern "C" void kernel_launch(void* const* d_in, const int* in_sizes, int n_in,
                              void* d_out, int out_size, void* d_ws, size_t ws_size,
                              hipStream_t stream) {
    (void)in_sizes; (void)n_in; (void)out_size; (void)ws_size;
    const float* pred = (const float*)d_in[0];
    const float* targ = (const float*)d_in[1];
    float* partials   = (float*)d_ws;          // 3 * 4096 floats = 48 KB
    float* out        = (float*)d_out;

    dim3 grid(NBX, NBY);
    loss_tile_kernel<<<grid, 256, 0, stream>>>(pred, targ, partials);
    loss_finalize_kernel<<<1, 32, 0, stream>>>(partials, out);
}